// ConcreteSelector_35768487641420
// MI455X (gfx1250) — compile-verified
//
#include <hip/hip_runtime.h>

typedef __attribute__((ext_vector_type(2))) float v2f;
typedef __attribute__((ext_vector_type(4))) float v4f;
typedef __attribute__((ext_vector_type(8))) float v8f;

#define KCH 32          // k channels
#define IS  10000       // input size
#define TI  128         // i-chunk per LDS tile
#define LDP 132         // padded LDS row stride (dwords): 132%64==4 -> conflict-free b64 A reads

// One block = one n-row, 64 threads = 2 waves (wave w owns k in [16w, 16w+16)).
__global__ __launch_bounds__(64) void concrete_selector_kernel(
    const float* __restrict__ x, const float* __restrict__ u,
    const float* __restrict__ logits, float* __restrict__ out, int n_rows)
{
    __shared__ float e_s[KCH][LDP];   // exp values, row-per-k
    __shared__ float b_s[2][TI];      // B table: row 0 = x chunk (numer), row 1 = ones (denom)

    const int n = blockIdx.x;
    if (n >= n_rows) return;                 // uniform per block
    const int t    = threadIdx.x;
    const int lane = t & 31;
    const int wave = t >> 5;                 // 0 or 1
    const int row  = lane & 15;              // A-row / B-col index within half-wave
    const bool hi  = lane >= 16;             // upper half: A cols +2, C rows M+8
    const int  kb  = wave * 16;

    const float* __restrict__ urow = u + (size_t)n * KCH * IS;
    const float* __restrict__ xrow = x + (size_t)n * IS;

    // ones row of the B table (written once; first use is after the phase-1 barrier)
    for (int j = t; j < TI; j += 64) b_s[1][j] = 1.0f;

    // per-lane operand base pointers for phase 2 (sel: cols 0-7 -> x, cols 8-15 -> 1.0)
    const float* aptr = &e_s[kb + row][hi ? 2 : 0];
    const float* bptr = &b_s[(row < 8) ? 0 : 1][hi ? 2 : 0];

    v8f c = {};  // C accumulator: cols 0-7 = numer, cols 8-15 = denom

    const float EPS  = 1e-10f;
    const float NLN2 = -0.69314718055994531f;   // -ln2
    const float CLG  = 0.43280851140899867f;    // LR_MULT/(TEMP*ln2) = 3/(10 ln2)

    for (int i0 = 0; i0 < IS; i0 += TI) {
        // ---- prefetch next chunk of the u stream (global_prefetch_b8) ----
        int ni0 = i0 + TI;
        if (ni0 < IS) {
            const float* pf = &urow[(size_t)(t & 31) * IS + ni0 + ((t >> 5) << 6)];
            __builtin_prefetch(pf, 0, 1);
        }

        const bool full = (i0 + TI) <= IS;       // uniform branch
        if (full) {
            // ---- phase 1 (fast): b128 non-temporal streaming, no guards ----
            for (int j = t; j < TI; j += 64) b_s[0][j] = xrow[i0 + j];
            #pragma unroll
            for (int p = 0; p < (KCH * TI) / (64 * 4); ++p) {   // 16 passes of 64 lanes x float4
                int flat = p * 256 + 4 * t;
                int k = flat >> 7;                // / TI
                int i = flat & (TI - 1);
                v4f uv = __builtin_nontemporal_load(
                            (const v4f*)(urow + (size_t)k * IS + i0 + i));
                v4f lg = *(const v4f*)(logits + (size_t)k * IS + i0 + i);
                v4f e4;
                #pragma unroll
                for (int q = 0; q < 4; ++q) {
                    float l1 = __builtin_amdgcn_logf(uv[q] + EPS);  // log2(u+eps)
                    float L  = fmaf(NLN2, l1, EPS);                 // -ln(u+eps)+eps
                    float l2 = __builtin_amdgcn_logf(L);            // log2(L)
                    float ea = fmaf(lg[q], CLG, -0.1f * l2);
                    e4[q] = __builtin_amdgcn_exp2f(ea);             // exp(score)
                }
                *(v4f*)&e_s[k][i] = e4;
            }
        } else {
            // ---- phase 1 (tail): guarded scalar, zero-fill pads ----
            for (int j = t; j < TI; j += 64) {
                int gi = i0 + j;
                b_s[0][j] = (gi < IS) ? xrow[gi] : 0.0f;
            }
            for (int idx = t; idx < KCH * TI; idx += 64) {
                int k  = idx >> 7;
                int i  = idx & (TI - 1);
                int gi = i0 + i;
                float e = 0.0f;
                if (gi < IS) {
                    float uv  = __builtin_nontemporal_load(&urow[(size_t)k * IS + gi]);
                    float lgv = logits[(size_t)k * IS + gi];
                    float l1  = __builtin_amdgcn_logf(uv + EPS);
                    float L   = fmaf(NLN2, l1, EPS);
                    float l2  = __builtin_amdgcn_logf(L);
                    float ea  = fmaf(lgv, CLG, -0.1f * l2);
                    e = __builtin_amdgcn_exp2f(ea);
                }
                e_s[k][i] = e;
            }
        }
        __syncthreads();

        // ---- phase 2: fp32 WMMA accumulate (EXEC all ones, no divergence) ----
        // A (16x4 f32): lane<16 -> K=ii,ii+1 of row `row`; lane>=16 -> K=ii+2,ii+3
        // B (4x16 f32): col `row`; cols 0-7 carry x[i], cols 8-15 carry 1.0
        #pragma unroll 8
        for (int ii = 0; ii < TI; ii += 4) {
            v2f a = *(const v2f*)(aptr + ii);
            v2f b = *(const v2f*)(bptr + ii);
            c = __builtin_amdgcn_wmma_f32_16x16x4_f32(
                    /*neg_a=*/false, a, /*neg_b=*/false, b,
                    /*c_mod=*/(short)0, c, /*reuse_a=*/false, /*reuse_b=*/false);
        }
        __syncthreads();
    }

    // ---- extract: lane N==0 divides numer by denom fetched from lane N==8 ----
    #pragma unroll
    for (int p = 0; p < 8; ++p) {
        float num = c[p];
        float den = __shfl_xor(num, 8, 32);     // col 0 <- col 8 (same M row)
        if (row == 0) {                          // lanes 0 (M=p) and 16 (M=p+8)
            int k = kb + p + (hi ? 8 : 0);
            out[(size_t)n * KCH + k] = num / den;
        }
    }
}

extern "C" void kernel_launch(void* const* d_in, const int* in_sizes, int n_in,
                              void* d_out, int out_size, void* d_ws, size_t ws_size,
                              hipStream_t stream) {
    const float* x      = (const float*)d_in[0];   // (N, 10000)
    const float* u      = (const float*)d_in[1];   // (N, 32, 10000)
    const float* logits = (const float*)d_in[2];   // (32, 10000)
    float* out = (float*)d_out;                    // (N, 32)
    int n_rows = in_sizes[0] / IS;
    concrete_selector_kernel<<<dim3(n_rows), dim3(64), 0, stream>>>(x, u, logits, out, n_rows);
}